// FirstArrivalTravelTime_20675972563826
// MI455X (gfx1250) — compile-verified
//
#include <hip/hip_runtime.h>
#include <hip/hip_bf16.h>
#include <math.h>

// ---------------- problem constants ----------------
#define NS    3000          // nsamples
#define NT    384           // ntraces
#define NC    3             // nchannels
#define NB    (NT*NC)       // 1152 batch signals
#define NSTA  101
#define NLTA  1001
#define NRES  600           // samples kept after pick
#define NCC   (2*NS-1)      // 5999 lags
#define NTILE 24            // 24*256 = 6144 >= 5999
#define GPS   (NTILE*256)   // 6144 padded gumbel stride
#define XPAD  32            // x padded 32 front/back
#define YPAD  256           // y padded 256 front/back
#define XPS   (NS + 2*XPAD) // 3064
#define YPS   (NS + 2*YPAD) // 3512

typedef float v2f __attribute__((ext_vector_type(2)));
typedef float v8f __attribute__((ext_vector_type(8)));

// ---------------- kernel 1: STA/LTA first-arrival picks ----------------
__global__ __launch_bounds__(256) void pick_kernel(const float* __restrict__ x,
                                                   const float* __restrict__ y,
                                                   float* __restrict__ arr_x,
                                                   float* __restrict__ arr_y) {
    int gid = blockIdx.x * blockDim.x + threadIdx.x;
    if (gid >= 2 * NB) return;
    const float* src = (gid < NB) ? x : y;
    float* dst       = (gid < NB) ? arr_x : arr_y;
    int b = (gid < NB) ? gid : gid - NB;

    double sta = 0.0, lta = 0.0;
    int pick = 0;
    for (int i = 0; i < NS; ++i) {
        float v = src[(size_t)i * NB + b];
        double cf = (double)v * (double)v;
        sta += cf; lta += cf;
        if (i >= NSTA) { float u = src[(size_t)(i - NSTA) * NB + b]; sta -= (double)u * (double)u; }
        if (i >= NLTA) { float u = src[(size_t)(i - NLTA) * NB + b]; lta -= (double)u * (double)u; }
        if (i >= NLTA) {
            double ratio = (sta / (double)NSTA) / (lta / (double)NLTA + 1e-12);
            if (ratio > 0.5) { pick = i; break; }   // first trigger (argmax of bool)
        }
    }
    dst[b] = (float)pick;
}

// ---------------- kernel 2: degree-15 polyfit (f64 normal eq + Cholesky) ----------------
__global__ __launch_bounds__(NT) void polyfit_kernel(const float* __restrict__ arr_x,
                                                     const float* __restrict__ arr_y,
                                                     float* __restrict__ thresh_x,
                                                     float* __restrict__ thresh_y) {
    __shared__ double scoef[16][6];
    int tid = threadIdx.x;

    if (tid == 0) {
        double mom[31];
        double rhs[16][6];
        for (int p = 0; p < 31; ++p) mom[p] = 0.0;
        for (int j = 0; j < 16; ++j) for (int c = 0; c < 6; ++c) rhs[j][c] = 0.0;

        const double half = 0.5 * (double)(NT - 1);
        for (int t = 0; t < NT; ++t) {
            double tn = ((double)t - half) / half;
            double a[6];
            for (int c = 0; c < 3; ++c) {
                a[c]     = (double)arr_x[t * 3 + c];
                a[3 + c] = (double)arr_y[t * 3 + c];
            }
            double pw = 1.0;
            for (int p = 0; p <= 30; ++p) {
                mom[p] += pw;
                if (p <= 15)
                    for (int c = 0; c < 6; ++c) rhs[p][c] += pw * a[c];
                pw *= tn;
            }
        }
        double G[16][16];
        for (int j = 0; j < 16; ++j)
            for (int k = 0; k < 16; ++k) G[j][k] = mom[j + k];
        for (int j = 0; j < 16; ++j) {
            for (int k = 0; k < j; ++k) {
                double s = G[j][k];
                for (int q = 0; q < k; ++q) s -= G[j][q] * G[k][q];
                G[j][k] = s / G[k][k];
            }
            double s = G[j][j];
            for (int q = 0; q < j; ++q) s -= G[j][q] * G[j][q];
            G[j][j] = sqrt(fmax(s, 1e-300));
        }
        for (int c = 0; c < 6; ++c) {
            double z[16];
            for (int j = 0; j < 16; ++j) {
                double s = rhs[j][c];
                for (int q = 0; q < j; ++q) s -= G[j][q] * z[q];
                z[j] = s / G[j][j];
            }
            for (int j = 15; j >= 0; --j) {
                double s = z[j];
                for (int q = j + 1; q < 16; ++q) s -= G[q][j] * scoef[q][c];
                scoef[j][c] = s / G[j][j];
            }
        }
    }
    __syncthreads();

    int t = tid;
    const double half = 0.5 * (double)(NT - 1);
    double tn = ((double)t - half) / half;
    double f[6] = {0, 0, 0, 0, 0, 0};
    double pw = 1.0;
    for (int j = 0; j < 16; ++j) {
        for (int c = 0; c < 6; ++c) f[c] += scoef[j][c] * pw;
        pw *= tn;
    }
    for (int c = 0; c < 3; ++c) {
        float fx = fminf(fmaxf((float)f[c],     0.f), (float)(NS - 1));
        float fy = fminf(fmaxf((float)f[3 + c], 0.f), (float)(NS - 1));
        thresh_x[t * 3 + c] = fx + (float)NRES;
        thresh_y[t * 3 + c] = fy + (float)NRES;
    }
}

// ---------------- kernel 3: zero padded scratch ----------------
__global__ void zero_kernel(float* __restrict__ p, size_t n) {
    size_t i = (size_t)blockIdx.x * blockDim.x + threadIdx.x;
    size_t stride = (size_t)gridDim.x * blockDim.x;
    for (; i < n; i += stride) p[i] = 0.f;
}

// ---------------- kernel 3b: pad gumbel to [NB, 6144] with -INF tail ----------------
__global__ __launch_bounds__(256) void gumbel_pad_kernel(const float* __restrict__ g,
                                                         float* __restrict__ gp) {
    size_t i = (size_t)blockIdx.x * blockDim.x + threadIdx.x;
    size_t n = (size_t)NB * GPS;
    size_t stride = (size_t)gridDim.x * blockDim.x;
    for (; i < n; i += stride) {
        int k = (int)(i % GPS);
        size_t b = i / GPS;
        gp[i] = (k < NCC) ? g[b * NCC + k] : -INFINITY;
    }
}

// ---------------- kernel 4: cut + LDS-tiled transpose [N,B] -> [B, pad+N+pad] ----------------
__global__ __launch_bounds__(256) void pack_kernel(const float* __restrict__ src,
                                                   const float* __restrict__ thresh,
                                                   float* __restrict__ dst,
                                                   int dstride, int doff) {
    __shared__ float tile[32][33];
    int n0 = blockIdx.x * 32;
    int b0 = blockIdx.y * 32;
    int tx = threadIdx.x & 31, ty = threadIdx.x >> 5;   // 32 x 8

    #pragma unroll
    for (int j = 0; j < 4; ++j) {
        int n = n0 + ty + 8 * j;
        int bb = b0 + tx;
        float v = 0.f;
        if (n < NS) v = src[(size_t)n * NB + bb];
        tile[ty + 8 * j][tx] = v;
    }
    __syncthreads();
    #pragma unroll
    for (int j = 0; j < 4; ++j) {
        int bb = b0 + ty + 8 * j;
        int n = n0 + tx;
        if (n < NS) {
            float v = tile[tx][ty + 8 * j];
            float th = thresh[bb];
            dst[(size_t)bb * dstride + doff + n] = ((float)n < th) ? v : 0.f;
        }
    }
}

// ---------------- kernel 5: WMMA xcorr + fused gumbel-argmax ----------------
// One block per batch signal; 8 waves x 3 lag-tiles of 256 lags. Each tile is a
// 16xKx16 GEMM via v_wmma_f32_16x16x4_f32 with scalar (SALU) loop control and
// K unrolled by 8 (two WMMAs/iter). Epilogue is branch-free: padded gumbel rows
// give 2x b128 loads per tile and cndmask-style argmax updates.
__global__ __launch_bounds__(256) void xcorr_argmax_kernel(const float* __restrict__ xp,
                                                           const float* __restrict__ yp,
                                                           const float* __restrict__ gpad,
                                                           float* __restrict__ contrib) {
    __shared__ float sx[XPS];
    __shared__ float sy[YPS];
    __shared__ float swmax[8];
    __shared__ float sbv[8];
    __shared__ int   sbi[8];

    int b   = blockIdx.x;
    int tid = threadIdx.x;
    int wave = __builtin_amdgcn_readfirstlane(tid >> 5);
    int lane = tid & 31;
    int h = lane >> 4;          // half-wave: K sub-pair / C row offset
    int m = lane & 15;          // A row / B column / C column

    // stage signals into LDS; fold |.|-max (validity) into the load
    float mx = 0.f;
    for (int i = tid; i < XPS; i += 256) {
        float v = xp[(size_t)b * XPS + i];
        sx[i] = v; mx = fmaxf(mx, fabsf(v));
    }
    for (int i = tid; i < YPS; i += 256) {
        float v = yp[(size_t)b * YPS + i];
        sy[i] = v; mx = fmaxf(mx, fabsf(v));
    }
    #pragma unroll
    for (int o = 16; o > 0; o >>= 1) mx = fmaxf(mx, __shfl_xor(mx, o, 32));
    if (lane == 0) swmax[wave] = mx;
    __syncthreads();

    float best = -INFINITY;
    int bestk = 0;

    for (int t = wave; t < NTILE; t += 8) {
        int k0 = t * 256;
        int l0 = k0 - (NS - 1);                       // lag base of this tile
        // scalar (wave-uniform) K bounds, multiple-of-8 trip count
        int pmin = __builtin_amdgcn_readfirstlane(max(0, -l0 - 15) & ~7);
        int plen = __builtin_amdgcn_readfirstlane(
                       (min(NS + 240, NS - l0) - pmin + 7) & ~7);

        // A[i][k] = x[l0 + i + p + k],  lane holds row i=m, K pair {2h, 2h+1}
        const float* ax = sx + XPAD + l0 + m + 2 * h + pmin;
        // B[k][n] = y[p + k - 16n],     lane holds col n=m, K pair {2h, 2h+1}
        const float* by = sy + YPAD - 16 * m + 2 * h + pmin;

        v8f acc = {};
        for (int p = 0; p < plen; p += 8) {
            v2f a0, b0, a1, b1;
            a0.x = ax[p];     a0.y = ax[p + 1];
            b0.x = by[p];     b0.y = by[p + 1];
            a1.x = ax[p + 4]; a1.y = ax[p + 5];
            b1.x = by[p + 4]; b1.y = by[p + 5];
            acc = __builtin_amdgcn_wmma_f32_16x16x4_f32(
                false, a0, false, b0, (short)0, acc, false, false);
            acc = __builtin_amdgcn_wmma_f32_16x16x4_f32(
                false, a1, false, b1, (short)0, acc, false, false);
        }

        // C element (M=v+8h, N=m) is cc index k0 + M + 16*N = (k0+16m+8h) + v:
        // the 8 gumbel values per lane are consecutive, 32B-aligned.
        int kb = k0 + 16 * m + 8 * h;
        const float4* gq = (const float4*)(gpad + (size_t)b * GPS + kb);
        float4 g0 = gq[0], g1 = gq[1];
        float gv[8] = {g0.x, g0.y, g0.z, g0.w, g1.x, g1.y, g1.z, g1.w};
        #pragma unroll
        for (int v = 0; v < 8; ++v) {
            float p = acc[v] * 1e6f + gv[v];
            int k = kb + v;
            bool better = (p > best) | ((p == best) & (k < bestk));
            best  = better ? p : best;
            bestk = better ? k : bestk;
        }
    }

    // wave32 argmax butterfly (first-index tie-break), branch-free
    #pragma unroll
    for (int o = 16; o > 0; o >>= 1) {
        float ov = __shfl_xor(best, o, 32);
        int   oi = __shfl_xor(bestk, o, 32);
        bool better = (ov > best) | ((ov == best) & (oi < bestk));
        best  = better ? ov : best;
        bestk = better ? oi : bestk;
    }
    if (lane == 0) { sbv[wave] = best; sbi[wave] = bestk; }
    __syncthreads();

    if (tid == 0) {
        float bv = sbv[0]; int bi = sbi[0];
        #pragma unroll
        for (int w = 1; w < 8; ++w) {
            bool better = (sbv[w] > bv) | ((sbv[w] == bv) & (sbi[w] < bi));
            bv = better ? sbv[w] : bv;
            bi = better ? sbi[w] : bi;
        }
        float gm = 0.f;
        #pragma unroll
        for (int w = 0; w < 8; ++w) gm = fmaxf(gm, swmax[w]);
        float c = (gm > 1e-5f) ? ((float)(bi - (NS - 1)) * 0.001f) : 0.f;
        contrib[b] = c;
    }
}

// ---------------- kernel 6: deterministic fixed-order final reduction ----------------
__global__ __launch_bounds__(256) void reduce_kernel(const float* __restrict__ contrib,
                                                     float* __restrict__ out) {
    __shared__ float red[256];
    int tid = threadIdx.x;
    float s = 0.f;
    for (int i = tid; i < NB; i += 256) s += contrib[i];
    red[tid] = s;
    __syncthreads();
    for (int o = 128; o > 0; o >>= 1) {
        if (tid < o) red[tid] = red[tid] + red[tid + o];
        __syncthreads();
    }
    if (tid == 0) out[0] = red[0];
}

// ---------------- launcher ----------------
extern "C" void kernel_launch(void* const* d_in, const int* in_sizes, int n_in,
                              void* d_out, int out_size, void* d_ws, size_t ws_size,
                              hipStream_t stream) {
    (void)in_sizes; (void)n_in; (void)out_size; (void)ws_size;

    const float* x      = (const float*)d_in[0];   // [3000,384,3]
    const float* y      = (const float*)d_in[1];   // [3000,384,3]
    const float* gumbel = (const float*)d_in[2];   // [1152,5999]
    float* out = (float*)d_out;

    float* ws = (float*)d_ws;
    float* arr_x    = ws;                       // [NB]
    float* arr_y    = arr_x + NB;               // [NB]
    float* thresh_x = arr_y + NB;               // [NB]
    float* thresh_y = thresh_x + NB;            // [NB]
    float* xp       = thresh_y + NB;            // [NB * XPS]
    float* yp       = xp + (size_t)NB * XPS;    // [NB * YPS]
    float* gpad     = yp + (size_t)NB * YPS;    // [NB * GPS], 16B-aligned
    float* contrib  = gpad + (size_t)NB * GPS;  // [NB]

    // 1) picks
    pick_kernel<<<(2 * NB + 255) / 256, 256, 0, stream>>>(x, y, arr_x, arr_y);
    // 2) polyfit -> cut thresholds
    polyfit_kernel<<<1, NT, 0, stream>>>(arr_x, arr_y, thresh_x, thresh_y);
    // 3) zero padded signal scratch; pad gumbel rows with -INF
    size_t nz = (size_t)NB * XPS + (size_t)NB * YPS;
    zero_kernel<<<2048, 256, 0, stream>>>(xp, nz);
    gumbel_pad_kernel<<<2048, 256, 0, stream>>>(gumbel, gpad);
    // 4) cut + transpose into padded [B,N] layouts
    dim3 pgrid((NS + 31) / 32, NB / 32);
    pack_kernel<<<pgrid, 256, 0, stream>>>(x, thresh_x, xp, XPS, XPAD);
    pack_kernel<<<pgrid, 256, 0, stream>>>(y, thresh_y, yp, YPS, YPAD);
    // 5) WMMA cross-correlation + fused gumbel argmax
    xcorr_argmax_kernel<<<NB, 256, 0, stream>>>(xp, yp, gpad, contrib);
    // 6) deterministic final sum
    reduce_kernel<<<1, 256, 0, stream>>>(contrib, out);
}